// GALExtrapolation_55198919688665
// MI455X (gfx1250) — compile-verified
//
#include <hip/hip_runtime.h>

// Problem dimensions (fixed by the reference)
constexpr int B_    = 4;
constexpr int T_    = 12;
constexpr int N_    = 5000;
constexpr int C_    = 16;
constexpr int H_    = 4;
constexpr int K_    = 9;
constexpr int TOUT_ = 12;           // 24 - T
constexpr int TH_   = T_ * H_;      // 48 (K-dim of final GEMM)
constexpr int M_    = B_ * N_ * C_; // 320000 rows of final GEMM (20000 tiles of 16)

typedef float v2f __attribute__((ext_vector_type(2)));
typedef float v8f __attribute__((ext_vector_type(8)));

__device__ __forceinline__ float selu_f(float x) {
    const float scale = 1.0507009873554805f;
    const float alpha = 1.6732632423543772f;
    return x > 0.0f ? scale * x : scale * alpha * (expf(x) - 1.0f);
}

// ---------------------------------------------------------------------------
// Stage 1: agg1[b,t,n,h,c] = selu(blend_t( sum_k W1[h,k]*x[b,t,nn[n,k],c] + b1[h] ))
// One thread per (b,n,h,c); loops T internally so the temporal blend is local.
// Neighbor index N means "zero pad": clamp address + multiply by 0/1 mask so
// the gather loads stay unconditional (no exec-predicated load blocks).
// ---------------------------------------------------------------------------
__global__ __launch_bounds__(256)
void stageA_kernel(const float* __restrict__ x, const int* __restrict__ nn,
                   const float* __restrict__ W1, const float* __restrict__ b1,
                   float* __restrict__ agg1) {
    int gid = blockIdx.x * blockDim.x + threadIdx.x;
    if (gid >= B_ * N_ * H_ * C_) return;
    int c   = gid & (C_ - 1);
    int h   = (gid >> 4) & (H_ - 1);
    int bn  = gid >> 6;          // b*N + n
    int n   = bn % N_;
    int b   = bn / N_;

    int   idx[K_];
    float w[K_];
#pragma unroll
    for (int k = 0; k < K_; ++k) {
        int id = nn[n * K_ + k];
        float msk = (id < N_) ? 1.0f : 0.0f;     // id == N -> zero pad row
        idx[k] = (id < N_) ? id : 0;             // clamp: address always valid
        w[k]   = W1[h * K_ + k] * msk;           // fold mask into the weight
    }
    float bias = b1[h];

    float prev = 0.0f;
    for (int t = 0; t < T_; ++t) {
        const float* xbt = x + (size_t)(b * T_ + t) * N_ * C_;
        float s = bias;
#pragma unroll
        for (int k = 0; k < K_; ++k) {
            s = fmaf(w[k], xbt[idx[k] * C_ + c], s);
        }
        float bl = (t == 0) ? s : 0.8f * s + 0.2f * prev;   // blend on pre-activation
        prev = s;
        agg1[(((size_t)(b * T_ + t) * N_ + n) * H_ + h) * C_ + c] = selu_f(bl);
    }
}

// ---------------------------------------------------------------------------
// Stage 2: agg2 = selu(blend_t( sum_k W2[k]*agg1[b,t,nn[n,k],h,c] + b2 ))
// Written directly in the (B*N*C, 48) layout the GEMM wants: j = t*H + h.
// ---------------------------------------------------------------------------
__global__ __launch_bounds__(256)
void stageB_kernel(const float* __restrict__ agg1, const int* __restrict__ nn,
                   const float* __restrict__ W2, const float* __restrict__ b2,
                   float* __restrict__ a_mat) {
    int gid = blockIdx.x * blockDim.x + threadIdx.x;
    if (gid >= B_ * N_ * H_ * C_) return;
    int c   = gid & (C_ - 1);
    int h   = (gid >> 4) & (H_ - 1);
    int bn  = gid >> 6;
    int n   = bn % N_;
    int b   = bn / N_;

    int   idx[K_];
    float w[K_];
#pragma unroll
    for (int k = 0; k < K_; ++k) {
        int id = nn[n * K_ + k];
        float msk = (id < N_) ? 1.0f : 0.0f;
        idx[k] = (id < N_) ? id : 0;
        w[k]   = W2[k] * msk;
    }
    float bias = b2[0];

    size_t row_base = ((size_t)(b * N_ + n) * C_ + c) * TH_;
    float prev = 0.0f;
    for (int t = 0; t < T_; ++t) {
        const float* abt = agg1 + (size_t)(b * T_ + t) * N_ * H_ * C_;
        float s = bias;
#pragma unroll
        for (int k = 0; k < K_; ++k) {
            s = fmaf(w[k], abt[(idx[k] * H_ + h) * C_ + c], s);
        }
        float bl = (t == 0) ? s : 0.8f * s + 0.2f * prev;
        prev = s;
        a_mat[row_base + t * H_ + h] = selu_f(bl);
    }
}

// ---------------------------------------------------------------------------
// Copy x into out[:, 0:12, :, :]   (out is (B, 24, N, C))
// ---------------------------------------------------------------------------
__global__ __launch_bounds__(256)
void copyx_kernel(const float* __restrict__ x, float* __restrict__ out) {
    int gid = blockIdx.x * blockDim.x + threadIdx.x;
    if (gid >= B_ * T_ * N_ * C_) return;
    int per_b = T_ * N_ * C_;
    int b = gid / per_b;
    int r = gid - b * per_b;
    out[(size_t)b * (2 * T_) * N_ * C_ + r] = x[gid];
}

// ---------------------------------------------------------------------------
// Stage 3: y = selu(A @ Ws^T + bs), A is (320000, 48), Ws is (12, 48).
// V_WMMA_F32_16X16X4_F32: one wave per 16-row tile, K looped 12x (48/4),
// B-matrix columns 12..15 zero-padded via clamp+mask (no exec-predicated
// loads), and all 12 B fragments hoisted out of the K loop — the inner loop
// is pure global_load_b64(A) + v_wmma. Exact grid fit -> EXEC all-1s.
//
// ISA layouts (7.12.2):
//   A 16x4 f32 : lane L (0-15): v0=A[M=L][K=0], v1=A[M=L][K=1];
//                lane L+16    : v0=A[M=L][K=2], v1=A[M=L][K=3]
//   B 4x16 f32 : lane L (0-15): v0=B[K=0][N=L], v1=B[K=1][N=L];
//                lane L+16    : v0=B[K=2][N=L], v1=B[K=3][N=L]
//   D 16x16 f32: vgpr r, lanes 0-15: D[M=r][N=lane]; lanes 16-31: D[M=r+8][N=lane-16]
// ---------------------------------------------------------------------------
__global__ __launch_bounds__(256)
void gemm_wmma_kernel(const float* __restrict__ a_mat, const float* __restrict__ Ws,
                      const float* __restrict__ bs, float* __restrict__ out) {
    int wave = (int)(blockIdx.x * (blockDim.x >> 5)) + (threadIdx.x >> 5); // tile id
    int lane = threadIdx.x & 31;
    int m0   = lane & 15;     // M within tile (also N column for B/D)
    int half = lane >> 4;     // 0: K=0,1 ; 1: K=2,3

    // ---- hoisted B fragments (loop-invariant; Ws is 12x48, L2-resident) ----
    int   colc = (m0 < TOUT_) ? m0 : 0;          // clamped column: address valid
    float msk  = (m0 < TOUT_) ? 1.0f : 0.0f;     // columns 12..15 are zero
    const float* wrow = Ws + colc * TH_;
    v2f Bfr[TH_ / 4];
#pragma unroll
    for (int kk = 0; kk < TH_; kk += 4) {
        int kr = kk + half * 2;
        Bfr[kk >> 2].x = wrow[kr + 0] * msk;
        Bfr[kk >> 2].y = wrow[kr + 1] * msk;
    }

    int row = wave * 16 + m0; // global A row for this lane's A fragment
    const float* arow = a_mat + (size_t)row * TH_;

    v8f acc = {};
#pragma unroll
    for (int kk = 0; kk < TH_; kk += 4) {
        int kr = kk + half * 2;
        v2f Af;
        Af.x = arow[kr + 0];
        Af.y = arow[kr + 1];
        acc = __builtin_amdgcn_wmma_f32_16x16x4_f32(
            /*neg_a=*/false, Af, /*neg_b=*/false, Bfr[kk >> 2],
            /*c_mod=*/(short)0, acc, /*reuse_a=*/false, /*reuse_b=*/false);
    }

    // Store: lane's column is tout = m0; rows are r + half*8 within the tile.
    if (m0 < TOUT_) {
        float bv = bs[m0];
#pragma unroll
        for (int r = 0; r < 8; ++r) {
            int mrow = wave * 16 + r + half * 8;      // = (b*N + n)*C + c
            int c  = mrow & (C_ - 1);
            int bn = mrow >> 4;
            int n  = bn % N_;
            int b  = bn / N_;
            float v = selu_f(acc[r] + bv);
            out[(((size_t)b * (2 * T_) + (T_ + m0)) * N_ + n) * C_ + c] = v;
        }
    }
}

extern "C" void kernel_launch(void* const* d_in, const int* in_sizes, int n_in,
                              void* d_out, int out_size, void* d_ws, size_t ws_size,
                              hipStream_t stream) {
    const float* x  = (const float*)d_in[0];
    const int*   nn = (const int*)  d_in[1];
    const float* W1 = (const float*)d_in[2];
    const float* b1 = (const float*)d_in[3];
    const float* W2 = (const float*)d_in[4];
    const float* b2 = (const float*)d_in[5];
    const float* Ws = (const float*)d_in[6];
    const float* bs = (const float*)d_in[7];
    float* out = (float*)d_out;

    float* agg1  = (float*)d_ws;                               // B*T*N*H*C floats
    float* a_mat = agg1 + (size_t)B_ * T_ * N_ * H_ * C_;      // B*N*C*48 floats

    int nthA = B_ * N_ * H_ * C_;                  // 1,280,000
    stageA_kernel<<<(nthA + 255) / 256, 256, 0, stream>>>(x, nn, W1, b1, agg1);
    stageB_kernel<<<(nthA + 255) / 256, 256, 0, stream>>>(agg1, nn, W2, b2, a_mat);

    int nCopy = B_ * T_ * N_ * C_;                 // 3,840,000
    copyx_kernel<<<(nCopy + 255) / 256, 256, 0, stream>>>(x, out);

    int tiles = M_ / 16;                           // 20000, exact
    gemm_wmma_kernel<<<tiles / 8, 256, 0, stream>>>(a_mat, Ws, bs, out); // 8 waves/block
}